// Ridge_62294205661210
// MI455X (gfx1250) — compile-verified
//
#include <hip/hip_runtime.h>
#include <hip/hip_bf16.h>
#include <stdint.h>

// Blocked recursive-least-squares (chunked Sherman-Morrison / Woodbury) for
// in-context ridge regression on gfx1250.
//
//   state: P = (lam*I + X_{<i}^T X_{<i})^{-1}   (64x64, LDS)
//          w = P * X^T y                         (64,    LDS)
//   per 16-point chunk:
//     V  = Xc @ P                (WMMA, 16x64x64)
//     C  = I + V @ Xc^T          (WMMA, 16x16x64)
//     LDL:  C = L D L^T          (tiny, LDS)
//     e  = L^{-1} (y - Xc w)  -> preds = y - e   (innovation identity)
//     W  = L^{-1} V ;  P -= W^T D^{-1} W         (WMMA rank-16 update, 64x64x16)
//     w += V^T (C^{-1} r)
//
// One workgroup per batch, 128 threads (4 wave32). The next chunk's 16x64
// tile is prefetched into a double-buffered LDS slot with
// GLOBAL_LOAD_ASYNC_TO_LDS_B128 (ASYNCcnt), overlapping the whole serial
// LDL/solve pipeline of the current chunk.

typedef __attribute__((ext_vector_type(2))) float v2f;
typedef __attribute__((ext_vector_type(8))) float v8f;

#define RR_BATCH 64
#define RR_NPTS  512
#define RR_D     64
#define RR_CH    16
#define RR_NCH   (RR_NPTS / RR_CH)
#define RR_TPB   128
#define RR_LAM   1.0f
#define RR_PAD   4

__global__ __launch_bounds__(RR_TPB)
void ridge_rls_wmma_kernel(const float* __restrict__ data,
                           const float* __restrict__ targets,
                           float* __restrict__ out) {
  __shared__ float sP [RR_D ][RR_D + RR_PAD];     // inverse Gram (symmetric)
  __shared__ float sXc[2][RR_CH][RR_D + RR_PAD];  // chunk tiles, double-buffered
  __shared__ float sV [RR_CH][RR_D + RR_PAD];     // Xc @ P
  __shared__ float sW [RR_CH][RR_D + RR_PAD];     // L^{-1} V
  __shared__ float sWd[RR_CH][RR_D + RR_PAD];     // -W / D   (negation folded in)
  __shared__ float sC [RR_CH][RR_CH + 1];         // I + V Xc^T, destroyed by LDL
  __shared__ float sL [RR_CH][RR_CH + 1];         // unit-lower multipliers
  __shared__ float sInvD[RR_CH];
  __shared__ float sy[RR_CH], sr[RR_CH], se[RR_CH], scv[RR_CH];
  __shared__ float sw[RR_D];

  const int b    = blockIdx.x;
  const int tid  = threadIdx.x;
  const int wid  = tid >> 5;
  const int lane = tid & 31;
  const int half = lane >> 4;    // 0: lanes 0-15, 1: lanes 16-31
  const int n16  = lane & 15;

  // ---- init: P = (1/lam) I, w = 0 ----
  for (int idx = tid; idx < RR_D * RR_D; idx += RR_TPB) {
    int r = idx >> 6, c = idx & 63;
    sP[r][c] = (r == c) ? (1.0f / RR_LAM) : 0.0f;
  }
  if (tid < RR_D) sw[tid] = 0.0f;

  const float* Xb = data    + (size_t)b * RR_NPTS * RR_D;
  const float* yb = targets + (size_t)b * RR_NPTS;
  float*       ob = out     + (size_t)b * RR_NPTS;

  // ---- preload chunk 0 synchronously into buffer 0 ----
  for (int idx = tid; idx < RR_CH * RR_D; idx += RR_TPB)
    sXc[0][idx >> 6][idx & 63] = Xb[idx];

  for (int chunk = 0; chunk < RR_NCH; ++chunk) {
    const int i0  = chunk * RR_CH;
    const int buf = chunk & 1;

    if (tid < RR_CH) sy[tid] = yb[i0 + tid];
    if (chunk > 0) {
      // drain last iteration's async prefetch into sXc[buf]
      asm volatile("s_wait_asynccnt 0x0" ::: "memory");
    }
    __syncthreads();   // sXc[buf] + sy ready for everyone

    // ---- async prefetch of next chunk into the alternate buffer ----
    // 16 rows x 256B; each B128 moves 16B/lane. 256 segments over 128 lanes.
    if (chunk + 1 < RR_NCH) {
      const float* src  = Xb + (size_t)(i0 + RR_CH) * RR_D;
      const unsigned lb = (unsigned)(uintptr_t)&sXc[buf ^ 1][0][0];
      for (int s = tid; s < RR_CH * (RR_D / 4); s += RR_TPB) {
        const int row = s >> 4, seg = s & 15;
        unsigned lds_addr = lb + (unsigned)((row * (RR_D + RR_PAD) + seg * 4) * 4);
        uint64_t gaddr = (uint64_t)(uintptr_t)(src + row * RR_D + seg * 4);
        asm volatile("global_load_async_to_lds_b128 %0, %1, off"
                     :: "v"(lds_addr), "v"(gaddr) : "memory");
      }
    }

    // ---- r = y - Xc @ w  (16 row dots on wave0 lanes; overlaps with WMMA) ----
    if (tid < RR_CH) {
      float acc = 0.0f;
      #pragma unroll 8
      for (int k = 0; k < RR_D; ++k) acc = fmaf(sXc[buf][tid][k], sw[k], acc);
      sr[tid] = sy[tid] - acc;
    }

    // ---- phase 1: V = Xc @ P ; wave 'wid' owns N-tile 'wid' ----
    {
      const int nt = wid;
      v8f acc = {};
      #pragma unroll
      for (int kb = 0; kb < RR_D; kb += 4) {
        v2f a, bb;
        a.x  = sXc[buf][n16][kb + 2 * half + 0];
        a.y  = sXc[buf][n16][kb + 2 * half + 1];
        bb.x = sP[kb + 2 * half + 0][nt * 16 + n16];
        bb.y = sP[kb + 2 * half + 1][nt * 16 + n16];
        acc = __builtin_amdgcn_wmma_f32_16x16x4_f32(false, a, false, bb,
                                                    (short)0, acc, false, false);
      }
      #pragma unroll
      for (int v = 0; v < 8; ++v)
        sV[v + 8 * half][nt * 16 + n16] = acc[v];
    }
    __syncthreads();

    // ---- phase 2: C = I + V @ Xc^T  (single tile, wave 0) ----
    if (wid == 0) {
      v8f acc = {};
      #pragma unroll
      for (int kb = 0; kb < RR_D; kb += 4) {
        v2f a, bb;
        a.x  = sV[n16][kb + 2 * half + 0];
        a.y  = sV[n16][kb + 2 * half + 1];
        bb.x = sXc[buf][n16][kb + 2 * half + 0];   // (Xc^T)[k][n] = Xc[n][k]
        bb.y = sXc[buf][n16][kb + 2 * half + 1];
        acc = __builtin_amdgcn_wmma_f32_16x16x4_f32(false, a, false, bb,
                                                    (short)0, acc, false, false);
      }
      #pragma unroll
      for (int v = 0; v < 8; ++v) {
        int rr = v + 8 * half;
        sC[rr][n16] = acc[v] + ((rr == n16) ? 1.0f : 0.0f);
      }
    }
    __syncthreads();

    // ---- phase 3: LDL^T of C (one barrier per pivot; row j frozen at step j) --
    for (int j = 0; j < RR_CH; ++j) {
      if (tid == 0) sInvD[j] = 1.0f / sC[j][j];   // frozen pivot, race-free read
      if (tid < RR_CH && tid > j) {
        float m = sC[tid][j] * (1.0f / sC[j][j]);
        sL[tid][j] = m;
        for (int k = j; k < RR_CH; ++k) sC[tid][k] -= m * sC[j][k];
      }
      __syncthreads();
    }

    // ---- phase 4: innovations e = L^{-1} r -> preds; c = L^{-T} D^{-1} e ----
    if (tid == 0) {
      for (int j = 0; j < RR_CH; ++j) {
        float e = sr[j];
        for (int m = 0; m < j; ++m) e -= sL[j][m] * se[m];
        se[j] = e;
        ob[i0 + j] = sy[j] - e;          // pred_i = x_i^T w_{prefix<i}
      }
      for (int j = RR_CH - 1; j >= 0; --j) {
        float z = se[j] * sInvD[j];
        for (int m = j + 1; m < RR_CH; ++m) z -= sL[m][j] * scv[m];
        scv[j] = z;
      }
    }
    __syncthreads();

    if (chunk == RR_NCH - 1) break;      // last chunk: state update not needed

    // ---- phase 5: W = L^{-1} V (column-independent fwd solve); Wd = -W/D ----
    if (tid < RR_D) {
      const int k = tid;
      for (int j = 0; j < RR_CH; ++j) {
        float wv = sV[j][k];
        for (int m = 0; m < j; ++m) wv -= sL[j][m] * sW[m][k];
        sW[j][k]  = wv;
        sWd[j][k] = -wv * sInvD[j];
      }
    }
    __syncthreads();

    // ---- phase 6: P += W^T @ (-D^{-1} W)   (rank-16 WMMA update, 16 tiles) ----
    for (int t = wid; t < 16; t += 4) {
      const int ti = t >> 2, tj = t & 3;
      v8f acc;
      #pragma unroll
      for (int v = 0; v < 8; ++v)
        acc[v] = sP[ti * 16 + v + 8 * half][tj * 16 + n16];
      #pragma unroll
      for (int kb = 0; kb < RR_CH; kb += 4) {
        v2f a, bb;
        a.x  = sW [kb + 2 * half + 0][ti * 16 + n16];   // (W^T)[m][k] = W[k][m]
        a.y  = sW [kb + 2 * half + 1][ti * 16 + n16];
        bb.x = sWd[kb + 2 * half + 0][tj * 16 + n16];
        bb.y = sWd[kb + 2 * half + 1][tj * 16 + n16];
        acc = __builtin_amdgcn_wmma_f32_16x16x4_f32(false, a, false, bb,
                                                    (short)0, acc, false, false);
      }
      #pragma unroll
      for (int v = 0; v < 8; ++v)
        sP[ti * 16 + v + 8 * half][tj * 16 + n16] = acc[v];
    }

    // ---- phase 7: w += V^T c ----
    if (tid < RR_D) {
      float acc = sw[tid];
      #pragma unroll
      for (int j = 0; j < RR_CH; ++j) acc = fmaf(sV[j][tid], scv[j], acc);
      sw[tid] = acc;
    }
    __syncthreads();
  }
}

extern "C" void kernel_launch(void* const* d_in, const int* in_sizes, int n_in,
                              void* d_out, int out_size, void* d_ws, size_t ws_size,
                              hipStream_t stream) {
  (void)in_sizes; (void)n_in; (void)d_ws; (void)ws_size; (void)out_size;
  const float* data    = (const float*)d_in[0];   // (64, 512, 64) f32
  const float* targets = (const float*)d_in[1];   // (64, 512)     f32
  float*       out     = (float*)d_out;           // (64, 512)     f32
  ridge_rls_wmma_kernel<<<dim3(RR_BATCH), dim3(RR_TPB), 0, stream>>>(data, targets, out);
}